// SupplyGraphModel_41549513621816
// MI455X (gfx1250) — compile-verified
//
#include <hip/hip_runtime.h>

typedef __attribute__((ext_vector_type(2))) float v2f;
typedef __attribute__((ext_vector_type(8))) float v8f;

#define NN 50000
#define NE 800000

// ---------------- zero workspace (msum0, msum1, cnt0, cnt1) ----------------
__global__ __launch_bounds__(256) void zero_kernel(float4* __restrict__ p, int n4) {
  int i = blockIdx.x * blockDim.x + threadIdx.x;
  if (i < n4) p[i] = make_float4(0.f, 0.f, 0.f, 0.f);
}

// ---------------- scatter-add: msum[dst] += h[src]; cnt[dst] += 1 ----------
// 16 threads per edge, each handling a float4 chunk of the 64-wide feature.
__global__ __launch_bounds__(256) void scatter_kernel(
    const float* __restrict__ h, const int* __restrict__ ei,
    float* __restrict__ msum, float* __restrict__ cnt) {
  long i = (long)blockIdx.x * blockDim.x + threadIdx.x;
  if (i >= (long)NE * 16) return;
  int e = (int)(i >> 4);
  int c = (int)(i & 15);
  int src = ei[e];        // ei[0][e]
  int dst = ei[NE + e];   // ei[1][e]
  const float4 v = *(const float4*)(h + src * 64 + c * 4);
  float* o = msum + dst * 64 + c * 4;
  atomicAdd(o + 0, v.x);
  atomicAdd(o + 1, v.y);
  atomicAdd(o + 2, v.z);
  atomicAdd(o + 3, v.w);
  if (c == 0) atomicAdd(cnt + dst, 1.0f);
}

// ---------------- msum -> mean (divide by max(cnt,1)) ----------------------
__global__ __launch_bounds__(256) void normalize_kernel(
    float* __restrict__ m0, float* __restrict__ m1,
    const float* __restrict__ c0, const float* __restrict__ c1) {
  int i = blockIdx.x * blockDim.x + threadIdx.x;
  if (i >= NN * 16) return;
  int n = i >> 4, c = i & 15;
  float i0 = 1.0f / fmaxf(c0[n], 1.0f);
  float i1 = 1.0f / fmaxf(c1[n], 1.0f);
  float4* p0 = (float4*)(m0 + n * 64 + c * 4);
  float4* p1 = (float4*)(m1 + n * 64 + c * 4);
  float4 a = *p0, b = *p1;
  a.x *= i0; a.y *= i0; a.z *= i0; a.w *= i0;
  b.x *= i1; b.y *= i1; b.z *= i1; b.w *= i1;
  *p0 = a; *p1 = b;
}

// ---------------- fused layer GEMM --------------------------------------
// out[N,64] = relu([mean0|mean1|h] (N,192) @ [Wl0;Wl1;Wr0+Wr1] (192,64) + bias)
// One 16x64 output tile per wave via V_WMMA_F32_16X16X4_F32.
// B staged in LDS as (k-pair, col) float2 with padded stride 66 -> ds_load_b64
// per fragment, bank-conflict free across the two K-halves of the wave.
__global__ __launch_bounds__(256) void sage_gemm_kernel(
    const float* __restrict__ mean0, const float* __restrict__ mean1,
    const float* __restrict__ h,
    const float* __restrict__ Wl,   // this layer's Wl[2,64,64]
    const float* __restrict__ Wr,   // this layer's Wr[2,64,64]
    const float* __restrict__ bl,   // this layer's bl[2,64]
    float* __restrict__ hout) {
  __shared__ float2 Bs[96 * 66];    // 192 k-rows as 96 pairs, 64 cols, pad 66
  for (int t = threadIdx.x; t < 192 * 64; t += 256) {
    int k = t >> 6, n = t & 63;
    float v;
    if (k < 64)        v = Wl[k * 64 + n];
    else if (k < 128)  v = Wl[64 * 64 + (k - 64) * 64 + n];
    else               v = Wr[(k - 128) * 64 + n] + Wr[64 * 64 + (k - 128) * 64 + n];
    ((float*)&Bs[(k >> 1) * 66 + n])[k & 1] = v;
  }
  __syncthreads();

  int wave = threadIdx.x >> 5;
  int lane = threadIdx.x & 31;
  int tile = blockIdx.x * 8 + wave;
  if (tile >= NN / 16) return;      // uniform per wave; EXEC stays all-ones

  int half = lane >> 4;             // K-half handled by this lane
  int m    = tile * 16 + (lane & 15);
  int n0   = lane & 15;
  const float* a0 = mean0 + m * 64 + half * 2;
  const float* a1 = mean1 + m * 64 + half * 2;
  const float* a2 = h     + m * 64 + half * 2;

  v8f acc0 = {}, acc1 = {}, acc2 = {}, acc3 = {};

#pragma unroll
  for (int s = 0; s < 16; ++s) {    // K block 0: mean0 (k = 0..63)
    v2f a = *(const v2f*)(a0 + 4 * s);
    const float2* br = &Bs[(2 * s + half) * 66];
    v2f b0 = *(const v2f*)(br + n0);
    v2f b1 = *(const v2f*)(br + n0 + 16);
    v2f b2 = *(const v2f*)(br + n0 + 32);
    v2f b3 = *(const v2f*)(br + n0 + 48);
    acc0 = __builtin_amdgcn_wmma_f32_16x16x4_f32(false, a, false, b0, (short)0, acc0, false, false);
    acc1 = __builtin_amdgcn_wmma_f32_16x16x4_f32(false, a, false, b1, (short)0, acc1, false, false);
    acc2 = __builtin_amdgcn_wmma_f32_16x16x4_f32(false, a, false, b2, (short)0, acc2, false, false);
    acc3 = __builtin_amdgcn_wmma_f32_16x16x4_f32(false, a, false, b3, (short)0, acc3, false, false);
  }
#pragma unroll
  for (int s = 0; s < 16; ++s) {    // K block 1: mean1 (k = 64..127)
    v2f a = *(const v2f*)(a1 + 4 * s);
    const float2* br = &Bs[(32 + 2 * s + half) * 66];
    v2f b0 = *(const v2f*)(br + n0);
    v2f b1 = *(const v2f*)(br + n0 + 16);
    v2f b2 = *(const v2f*)(br + n0 + 32);
    v2f b3 = *(const v2f*)(br + n0 + 48);
    acc0 = __builtin_amdgcn_wmma_f32_16x16x4_f32(false, a, false, b0, (short)0, acc0, false, false);
    acc1 = __builtin_amdgcn_wmma_f32_16x16x4_f32(false, a, false, b1, (short)0, acc1, false, false);
    acc2 = __builtin_amdgcn_wmma_f32_16x16x4_f32(false, a, false, b2, (short)0, acc2, false, false);
    acc3 = __builtin_amdgcn_wmma_f32_16x16x4_f32(false, a, false, b3, (short)0, acc3, false, false);
  }
#pragma unroll
  for (int s = 0; s < 16; ++s) {    // K block 2: h (k = 128..191)
    v2f a = *(const v2f*)(a2 + 4 * s);
    const float2* br = &Bs[(64 + 2 * s + half) * 66];
    v2f b0 = *(const v2f*)(br + n0);
    v2f b1 = *(const v2f*)(br + n0 + 16);
    v2f b2 = *(const v2f*)(br + n0 + 32);
    v2f b3 = *(const v2f*)(br + n0 + 48);
    acc0 = __builtin_amdgcn_wmma_f32_16x16x4_f32(false, a, false, b0, (short)0, acc0, false, false);
    acc1 = __builtin_amdgcn_wmma_f32_16x16x4_f32(false, a, false, b1, (short)0, acc1, false, false);
    acc2 = __builtin_amdgcn_wmma_f32_16x16x4_f32(false, a, false, b2, (short)0, acc2, false, false);
    acc3 = __builtin_amdgcn_wmma_f32_16x16x4_f32(false, a, false, b3, (short)0, acc3, false, false);
  }

  // Epilogue: bias (bl[0,:] + bl[1,:]) + ReLU. C layout: VGPR i -> row i / i+8.
  int rowBase = tile * 16 + half * 8;
#pragma unroll
  for (int i = 0; i < 8; ++i) {
    float* orow = hout + (rowBase + i) * 64;
    int ca = n0, cb = n0 + 16, cc = n0 + 32, cd = n0 + 48;
    orow[ca] = fmaxf(acc0[i] + bl[ca] + bl[64 + ca], 0.f);
    orow[cb] = fmaxf(acc1[i] + bl[cb] + bl[64 + cb], 0.f);
    orow[cc] = fmaxf(acc2[i] + bl[cc] + bl[64 + cc], 0.f);
    orow[cd] = fmaxf(acc3[i] + bl[cd] + bl[64 + cd], 0.f);
  }
}

// ---------------- final linear: out[N,32] = h2 @ lin_W + lin_b -------------
__global__ __launch_bounds__(256) void out_gemm_kernel(
    const float* __restrict__ h, const float* __restrict__ W,  // [64,32]
    const float* __restrict__ b, float* __restrict__ out) {
  __shared__ float2 Bs[32 * 33];
  for (int t = threadIdx.x; t < 64 * 32; t += 256) {
    int k = t >> 5, n = t & 31;
    ((float*)&Bs[(k >> 1) * 33 + n])[k & 1] = W[k * 32 + n];
  }
  __syncthreads();

  int wave = threadIdx.x >> 5;
  int lane = threadIdx.x & 31;
  int tile = blockIdx.x * 8 + wave;
  if (tile >= NN / 16) return;

  int half = lane >> 4;
  int m    = tile * 16 + (lane & 15);
  int n0   = lane & 15;
  const float* ar = h + m * 64 + half * 2;

  v8f acc0 = {}, acc1 = {};
#pragma unroll
  for (int s = 0; s < 16; ++s) {
    v2f a = *(const v2f*)(ar + 4 * s);
    const float2* br = &Bs[(2 * s + half) * 33];
    v2f b0 = *(const v2f*)(br + n0);
    v2f b1 = *(const v2f*)(br + n0 + 16);
    acc0 = __builtin_amdgcn_wmma_f32_16x16x4_f32(false, a, false, b0, (short)0, acc0, false, false);
    acc1 = __builtin_amdgcn_wmma_f32_16x16x4_f32(false, a, false, b1, (short)0, acc1, false, false);
  }

  int rowBase = tile * 16 + half * 8;
#pragma unroll
  for (int i = 0; i < 8; ++i) {
    float* orow = out + (rowBase + i) * 32;
    orow[n0]      = acc0[i] + b[n0];
    orow[n0 + 16] = acc1[i] + b[n0 + 16];
  }
}

extern "C" void kernel_launch(void* const* d_in, const int* in_sizes, int n_in,
                              void* d_out, int out_size, void* d_ws, size_t ws_size,
                              hipStream_t stream) {
  const float* x    = (const float*)d_in[0];   // [50000,64]
  const int*   ei0  = (const int*)d_in[1];     // [2,800000]
  const int*   ei1  = (const int*)d_in[2];     // [2,800000]
  const float* Wl   = (const float*)d_in[3];   // [2,2,64,64]
  const float* Wr   = (const float*)d_in[4];   // [2,2,64,64]
  const float* bl   = (const float*)d_in[5];   // [2,2,64]
  const float* linW = (const float*)d_in[6];   // [64,32]
  const float* linb = (const float*)d_in[7];   // [32]
  float* out = (float*)d_out;                  // [50000,32]

  float* ws    = (float*)d_ws;
  float* msum0 = ws;                 // 3,200,000 f
  float* msum1 = ws + 3200000;       // 3,200,000 f
  float* cnt0  = ws + 6400000;       //    50,000 f
  float* cnt1  = ws + 6450000;       //    50,000 f
  float* h1    = ws + 6500000;       // 3,200,000 f
  float* h2    = ws + 9700000;       // 3,200,000 f

  dim3 blk(256);
  const int ZN4 = 6500000 / 4;                       // zero msum0,msum1,cnt0,cnt1
  const int ZB  = (ZN4 + 255) / 256;
  const int SB  = (int)(((long)NE * 16 + 255) / 256);
  const int NB  = (NN * 16 + 255) / 256;
  const int GB  = (NN / 16 + 7) / 8;                 // 3125 tiles / 8 waves

  const float* hcur = x;
  float* houts[2] = {h1, h2};
  for (int l = 0; l < 2; ++l) {
    zero_kernel<<<ZB, blk, 0, stream>>>((float4*)ws, ZN4);
    scatter_kernel<<<SB, blk, 0, stream>>>(hcur, ei0, msum0, cnt0);
    scatter_kernel<<<SB, blk, 0, stream>>>(hcur, ei1, msum1, cnt1);
    normalize_kernel<<<NB, blk, 0, stream>>>(msum0, msum1, cnt0, cnt1);
    sage_gemm_kernel<<<GB, blk, 0, stream>>>(msum0, msum1, hcur,
                                             Wl + l * 2 * 64 * 64,
                                             Wr + l * 2 * 64 * 64,
                                             bl + l * 2 * 64,
                                             houts[l]);
    hcur = houts[l];
  }
  out_gemm_kernel<<<GB, blk, 0, stream>>>(h2, linW, linb, out);
}